// MpnnConv_24945170055591
// MI455X (gfx1250) — compile-verified
//
#include <hip/hip_runtime.h>

// Problem constants (match reference).
#define B_    8
#define N_    512
#define CIN   64
#define CE    16
#define CMID  64
#define COUT  64
#define NWAVES 8   // waves per workgroup in the edge kernel

typedef float v2f __attribute__((ext_vector_type(2)));
typedef float v8f __attribute__((ext_vector_type(8)));

// Exact-precision CDNA5 matrix op: D(16x16,f32) = A(16x4,f32) x B(4x16,f32) + C.
// Fragment mapping per ISA 7.12.2: element (row = lane&15, K = vgprIdx + 2*(lane>>4)).
static __device__ __forceinline__ v8f wmma_f32(v2f a, v2f b, v8f c) {
  return __builtin_amdgcn_wmma_f32_16x16x4_f32(false, a, false, b, (short)0, c,
                                               false, false);
}

// ---------------------------------------------------------------------------
// Kernel 1: per-node projections  M1 = x@W1^T + (b1+b2+be),  M2 = x@W2^T.
// One wave per 16-node tile; blockIdx.y selects which matrix.
// ---------------------------------------------------------------------------
__global__ __launch_bounds__(32, 1) void node_linear_kernel(
    const float* __restrict__ x,
    const float* __restrict__ W1, const float* __restrict__ b1,
    const float* __restrict__ W2, const float* __restrict__ b2,
    const float* __restrict__ be,
    float* __restrict__ M1, float* __restrict__ M2) {
  const int tile = blockIdx.x;          // 16-node tile, 0..255
  const int mat  = blockIdx.y;          // 0 -> M1 (W1 + fused biases), 1 -> M2 (W2)
  const int lane = threadIdx.x & 31;
  const int lo = lane & 15, hi = lane >> 4;
  const float* W = mat ? W2 : W1;
  float*       Y = mat ? M2 : M1;
  const int n0 = tile * 16;

  // A fragments: x tile [16 nodes x 64 in], 16 K-steps of 4.
  v2f a[16];
  const float* xrow = x + (size_t)(n0 + lo) * CIN;
#pragma unroll
  for (int ks = 0; ks < 16; ++ks)
    a[ks] = *(const v2f*)(xrow + ks * 4 + 2 * hi);

#pragma unroll
  for (int nt = 0; nt < 4; ++nt) {
    const int col = nt * 16 + lo;
    float bias = 0.0f;
    if (mat == 0) bias = b1[col] + b2[col] + be[col];  // fold b1+b2+be into M1
    v8f c;
#pragma unroll
    for (int v = 0; v < 8; ++v) c[v] = bias;
#pragma unroll
    for (int ks = 0; ks < 16; ++ks) {
      v2f bf = *(const v2f*)(W + (size_t)col * CIN + ks * 4 + 2 * hi);
      c = wmma_f32(a[ks], bf, c);
    }
#pragma unroll
    for (int v = 0; v < 8; ++v)
      Y[(size_t)(n0 + v + 8 * hi) * CMID + col] = c[v];
  }
}

// ---------------------------------------------------------------------------
// Kernel 2: heavy edge pass + fused tail.
// Block = (b, j-tile of 16). 8 waves stride the i dimension.
// Per i: D[j,m] = relu( edgeTile(16x16) @ We^T + M1[j,:] + M2[i,:] ),
//        acc[j,m] += adj[b,i,j] * D[j,m]; also deg[j] += adj[b,i,j].
// Tail (wave 0): agg = s@Wn^T + bn*deg; out = x@Wo1^T + agg@Wo2^T + bo1 + bo2.
// ---------------------------------------------------------------------------
__global__ __launch_bounds__(256, 1) void mpnn_edge_kernel(
    const float* __restrict__ xg,  const float* __restrict__ adj,
    const float* __restrict__ ea,  const float* __restrict__ We,
    const float* __restrict__ Wn,  const float* __restrict__ bn,
    const float* __restrict__ Wo1, const float* __restrict__ bo1,
    const float* __restrict__ Wo2, const float* __restrict__ bo2,
    const float* __restrict__ M1,  const float* __restrict__ M2,
    float* __restrict__ out) {
  __shared__ float red[NWAVES * 16 * CMID];  // 32 KB partial sums / scratch
  __shared__ float dred[NWAVES * 16];        // degree partials

  const int jt = blockIdx.x;           // j-tile 0..31
  const int b  = blockIdx.y;           // batch 0..7
  const int j0 = jt * 16;
  const int tid  = threadIdx.x;
  const int wave = tid >> 5;
  const int lane = tid & 31;
  const int lo = lane & 15, hi = lane >> 4;

  // B fragments of We^T (loop invariant): B[k,m] = We[m,k].
  v2f bw[4][4];
#pragma unroll
  for (int nt = 0; nt < 4; ++nt)
#pragma unroll
    for (int ks = 0; ks < 4; ++ks)
      bw[nt][ks] = *(const v2f*)(We + (size_t)(nt * 16 + lo) * CE + ks * 4 + 2 * hi);

  // Per-row C-init constants: m1t[nt][v] = M1[b, j0+v+8*hi, nt*16+lo].
  v8f m1t[4];
#pragma unroll
  for (int nt = 0; nt < 4; ++nt)
#pragma unroll
    for (int v = 0; v < 8; ++v)
      m1t[nt][v] = M1[((size_t)b * N_ + j0 + v + 8 * hi) * CMID + nt * 16 + lo];

  v8f acc[4];
#pragma unroll
  for (int nt = 0; nt < 4; ++nt)
#pragma unroll
    for (int v = 0; v < 8; ++v) acc[nt][v] = 0.0f;
  float dacc = 0.0f;

  for (int i = wave; i < N_; i += NWAVES) {
    const float* erow = ea + ((size_t)(b * N_ + i) * N_ + j0) * CE;
    if (i + NWAVES < N_)  // global_prefetch_b8 of next edge tile
      __builtin_prefetch(ea + ((size_t)(b * N_ + i + NWAVES) * N_ + j0) * CE + lane * 16, 0, 0);

    // A fragments: contiguous 16(j) x 16(c) edge tile.
    v2f a0 = *(const v2f*)(erow + lo * CE + 0  + 2 * hi);
    v2f a1 = *(const v2f*)(erow + lo * CE + 4  + 2 * hi);
    v2f a2 = *(const v2f*)(erow + lo * CE + 8  + 2 * hi);
    v2f a3 = *(const v2f*)(erow + lo * CE + 12 + 2 * hi);

    const float adjv = adj[((size_t)b * N_ + i) * N_ + j0 + lo];
    float m2v[4];
#pragma unroll
    for (int nt = 0; nt < 4; ++nt)
      m2v[nt] = M2[((size_t)b * N_ + i) * CMID + nt * 16 + lo];

    dacc += adjv;
    float ash[8];  // adj value for D row v+8*hi (held by lane v+8*hi)
#pragma unroll
    for (int v = 0; v < 8; ++v) ash[v] = __shfl(adjv, v + 8 * hi, 32);

#pragma unroll
    for (int nt = 0; nt < 4; ++nt) {
      v8f c;
#pragma unroll
      for (int v = 0; v < 8; ++v) c[v] = m1t[nt][v] + m2v[nt];
      c = wmma_f32(a0, bw[nt][0], c);
      c = wmma_f32(a1, bw[nt][1], c);
      c = wmma_f32(a2, bw[nt][2], c);
      c = wmma_f32(a3, bw[nt][3], c);
#pragma unroll
      for (int v = 0; v < 8; ++v)
        acc[nt][v] = fmaf(fmaxf(c[v], 0.0f), ash[v], acc[nt][v]);
    }
  }

  // Stash per-wave partials (layout [j][m]) and reduce across waves.
#pragma unroll
  for (int nt = 0; nt < 4; ++nt)
#pragma unroll
    for (int v = 0; v < 8; ++v)
      red[wave * (16 * CMID) + (v + 8 * hi) * CMID + nt * 16 + lo] = acc[nt][v];
  if (lane < 16) dred[wave * 16 + lane] = dacc;
  __syncthreads();

  for (int e = tid; e < 16 * CMID; e += 256) {
    float s = 0.0f;
#pragma unroll
    for (int w = 0; w < NWAVES; ++w) s += red[w * (16 * CMID) + e];
    red[e] = s;  // s[j][m] lives in red[0:1024]
  }
  if (tid < 16) {
    float s = 0.0f;
#pragma unroll
    for (int w = 0; w < NWAVES; ++w) s += dred[w * 16 + tid];
    dred[tid] = s;  // deg[j]
  }
  __syncthreads();
  if (wave != 0) return;

  // ------------------- tail: wave 0, all lanes active --------------------
  float* sbuf = red;                 // s[j][m]
  float* tbuf = red + 16 * CMID;     // agg[j][k]

  const float degv = dred[lo];
  float dsh[8];
#pragma unroll
  for (int v = 0; v < 8; ++v) dsh[v] = __shfl(degv, v + 8 * hi, 32);

  // agg = s @ Wn^T + bn * deg
#pragma unroll
  for (int nt = 0; nt < 4; ++nt) {
    const int col = nt * 16 + lo;
    const float bnv = bn[col];
    v8f c;
#pragma unroll
    for (int v = 0; v < 8; ++v) c[v] = bnv * dsh[v];
#pragma unroll
    for (int ks = 0; ks < 16; ++ks) {
      v2f a  = *(const v2f*)(sbuf + lo * CMID + ks * 4 + 2 * hi);
      v2f bf = *(const v2f*)(Wn + (size_t)col * CMID + ks * 4 + 2 * hi);
      c = wmma_f32(a, bf, c);
    }
#pragma unroll
    for (int v = 0; v < 8; ++v) tbuf[(v + 8 * hi) * CMID + col] = c[v];
  }

  // out = agg @ Wo2^T + x @ Wo1^T + (bo1 + bo2)   (single accumulator chain)
#pragma unroll
  for (int nt = 0; nt < 4; ++nt) {
    const int col = nt * 16 + lo;
    const float bias = bo1[col] + bo2[col];
    v8f c;
#pragma unroll
    for (int v = 0; v < 8; ++v) c[v] = bias;
#pragma unroll
    for (int ks = 0; ks < 16; ++ks) {
      v2f a  = *(const v2f*)(tbuf + lo * CMID + ks * 4 + 2 * hi);
      v2f bf = *(const v2f*)(Wo2 + (size_t)col * CMID + ks * 4 + 2 * hi);
      c = wmma_f32(a, bf, c);
    }
#pragma unroll
    for (int ks = 0; ks < 16; ++ks) {
      v2f a  = *(const v2f*)(xg + ((size_t)b * N_ + j0 + lo) * CIN + ks * 4 + 2 * hi);
      v2f bf = *(const v2f*)(Wo1 + (size_t)col * CIN + ks * 4 + 2 * hi);
      c = wmma_f32(a, bf, c);
    }
#pragma unroll
    for (int v = 0; v < 8; ++v)
      out[((size_t)b * N_ + j0 + v + 8 * hi) * COUT + col] = c[v];
  }
}

// ---------------------------------------------------------------------------
extern "C" void kernel_launch(void* const* d_in, const int* in_sizes, int n_in,
                              void* d_out, int out_size, void* d_ws, size_t ws_size,
                              hipStream_t stream) {
  const float* x   = (const float*)d_in[0];
  const float* adj = (const float*)d_in[1];
  const float* ea  = (const float*)d_in[2];
  const float* W1  = (const float*)d_in[3];
  const float* b1  = (const float*)d_in[4];
  const float* W2  = (const float*)d_in[5];
  const float* b2  = (const float*)d_in[6];
  const float* We  = (const float*)d_in[7];
  const float* be  = (const float*)d_in[8];
  const float* Wn  = (const float*)d_in[9];
  const float* bn  = (const float*)d_in[10];
  const float* Wo1 = (const float*)d_in[11];
  const float* bo1 = (const float*)d_in[12];
  const float* Wo2 = (const float*)d_in[13];
  const float* bo2 = (const float*)d_in[14];
  float* out = (float*)d_out;

  // Workspace: M1 and M2, each B*N*CMID floats (1 MB) -> needs 2 MB of d_ws.
  float* M1 = (float*)d_ws;
  float* M2 = M1 + (size_t)B_ * N_ * CMID;

  node_linear_kernel<<<dim3(B_ * N_ / 16, 2), 32, 0, stream>>>(
      x, W1, b1, W2, b2, be, M1, M2);

  mpnn_edge_kernel<<<dim3(N_ / 16, B_), 256, 0, stream>>>(
      x, adj, ea, We, Wn, bn, Wo1, bo1, Wo2, bo2, M1, M2, out);
}